// SS1D_block_23192823398802
// MI455X (gfx1250) — compile-verified
//
#include <hip/hip_runtime.h>

// Problem constants (from reference)
constexpr int kB = 4;
constexpr int kL = 4096;
constexpr int kD = 96;
constexpr int kN = 16;
constexpr int kR = 4;
constexpr int kK = 2;
constexpr int kC = kR + 2 * kN;   // 36 projection outputs
constexpr int kS = 32;            // scan segments per sequence
constexpr int kSeg = kL / kS;     // 128 steps per segment
constexpr int kWS = 98;           // padded LDS row stride for W (even + conflict-free)

typedef __attribute__((ext_vector_type(2))) float v2f;
typedef __attribute__((ext_vector_type(8))) float v8f;

__device__ __forceinline__ v8f wmma_f32_4(v2f a, v2f b, v8f c) {
  // D(16x16,f32) = A(16x4,f32) x B(4x16,f32) + C   -> v_wmma_f32_16x16x4_f32
  return __builtin_amdgcn_wmma_f32_16x16x4_f32(false, a, false, b, (short)0, c,
                                               false, false);
}

__device__ __forceinline__ float softplus_f(float v) {
  return (v > 20.f) ? v : log1pf(__expf(v));
}

// ---------------------------------------------------------------------------
// Kernel 1: fused projection.
//   x_dbl = xs @ W^T  (per k), split into dt_r / Bs / Cs
//   dt    = softplus(dt_r @ dtw^T + dtb)
// Both GEMMs on V_WMMA_F32_16X16X4_F32. Grid: (B*K, L/128), 256 thr (8 waves),
// each wave owns a 16-position tile.
// ---------------------------------------------------------------------------
__global__ __launch_bounds__(256) void ss1d_proj_kernel(
    const float* __restrict__ x, const float* __restrict__ xw,
    const float* __restrict__ dtw, const float* __restrict__ dtb,
    float* __restrict__ dt_ws, float* __restrict__ Bs_ws,
    float* __restrict__ Cs_ws) {
  __shared__ float sW[kC * kWS];       // 36 x 98   (x-projection weights)
  __shared__ float sDtw[kD * kR];      // 96 x 4    (dt projection weights)
  __shared__ float sDtb[kD];           // 96        (dt bias)
  __shared__ float sDtr[8 * 16 * kR];  // per-wave dt_r staging (16x4 tiles)

  const int tid = threadIdx.x;
  const int bk = blockIdx.x;
  const int b = bk >> 1;
  const int k = bk & 1;

  for (int i = tid; i < kC * kD; i += 256) {
    const int c = i / kD, d = i - c * kD;
    sW[c * kWS + d] = xw[k * kC * kD + i];
  }
  for (int i = tid; i < kD * kR; i += 256) sDtw[i] = dtw[k * kD * kR + i];
  for (int i = tid; i < kD; i += 256) sDtb[i] = dtb[k * kD + i];
  __syncthreads();

  const int wave = tid >> 5;
  const int lane = tid & 31;
  const int n = lane & 15;  // column (B/C/D layout) == row m (A layout)
  const int g = lane >> 4;  // lane-group
  const int pos0 = blockIdx.y * 128 + wave * 16;

  // A operand: row m == n of the 16-position tile; K-pair = d0+2g, d0+2g+1
  const int lA = pos0 + n;
  const int glA = k ? (kL - 1 - lA) : lA;
  const float* xrow = x + ((size_t)b * kL + glA) * kD + 2 * g;

  // --- GEMM 1: 24 K-steps over D=96, three 16-wide column tiles (36 cols) ---
  v8f acc[3] = {v8f{}, v8f{}, v8f{}};
  int c2 = 32 + n;
  if (c2 >= kC) c2 = 0;  // columns 36..47 are dead, clamp address
  const float* wr0 = &sW[(0 + n) * kWS + 2 * g];
  const float* wr1 = &sW[(16 + n) * kWS + 2 * g];
  const float* wr2 = &sW[c2 * kWS + 2 * g];
#pragma unroll
  for (int kk = 0; kk < kD / 4; ++kk) {
    const int d0 = kk * 4;
    const v2f a = *reinterpret_cast<const v2f*>(xrow + d0);
    const v2f b0 = *reinterpret_cast<const v2f*>(wr0 + d0);
    const v2f b1 = *reinterpret_cast<const v2f*>(wr1 + d0);
    const v2f b2 = *reinterpret_cast<const v2f*>(wr2 + d0);
    acc[0] = wmma_f32_4(a, b0, acc[0]);
    acc[1] = wmma_f32_4(a, b1, acc[1]);
    acc[2] = wmma_f32_4(a, b2, acc[2]);
  }

  // --- Scatter: dt_r -> LDS (for GEMM 2), Bs/Cs -> workspace ---
#pragma unroll
  for (int t = 0; t < 3; ++t) {
#pragma unroll
    for (int i = 0; i < 8; ++i) {
      const int row = i + 8 * g;          // position within tile
      const int l = pos0 + row;
      const int c = 16 * t + n;           // projection output index
      const float v = acc[t][i];
      if (c < kR) {
        sDtr[(wave * 16 + row) * kR + c] = v;
      } else if (c < kR + kN) {
        Bs_ws[((size_t)bk * kL + l) * kN + (c - kR)] = v;
      } else if (c < kC) {
        Cs_ws[((size_t)bk * kL + l) * kN + (c - kR - kN)] = v;
      }
    }
  }
  // sDtr producer/consumer are the same wave: LDS ops are in-order per wave.

  // --- GEMM 2: dt[16 pos x 96 d] = dt_r(16x4) x dtw^T(4x96), K=4 exact ---
  const v2f a2 =
      *reinterpret_cast<const v2f*>(&sDtr[(wave * 16 + n) * kR + 2 * g]);
#pragma unroll
  for (int t2 = 0; t2 < kD / 16; ++t2) {
    const int d0t = t2 * 16;
    const v2f bd =
        *reinterpret_cast<const v2f*>(&sDtw[(d0t + n) * kR + 2 * g]);
    v8f accD = {};
    accD = wmma_f32_4(a2, bd, accD);
#pragma unroll
    for (int i = 0; i < 8; ++i) {
      const int row = i + 8 * g;
      const int l = pos0 + row;
      const int d = d0t + n;
      const float v = softplus_f(accD[i] + sDtb[d]);
      dt_ws[((size_t)bk * kL + l) * kD + d] = v;
    }
  }
}

// ---------------------------------------------------------------------------
// Kernels 2a/2c: segmented diagonal linear recurrence.
//   FINAL=0: per segment compute P = prod(dA), Q = local scan end-state (h0=0)
//   FINAL=1: re-run segment with true incoming state, emit y
// Grid: B*K*S blocks, 96 threads (one d per thread), h[16]/a[16] in VGPRs.
// ---------------------------------------------------------------------------
template <int FINAL>
__global__ __launch_bounds__(96) void ss1d_scan_seg_kernel(
    const float* __restrict__ x, const float* __restrict__ A_logs,
    const float* __restrict__ Dsv, const float* __restrict__ dt_ws,
    const float* __restrict__ Bs_ws, const float* __restrict__ Cs_ws,
    const float* __restrict__ hpre_ws, float* __restrict__ P_ws,
    float* __restrict__ Q_ws, float* __restrict__ ys_ws) {
  const int bks = blockIdx.x;  // bk * kS + s
  const int bk = bks >> 5;
  const int s = bks & (kS - 1);
  const int b = bk >> 1;
  const int k = bk & 1;
  const int d = threadIdx.x;

  float a[kN], h[kN], P[kN];
#pragma unroll
  for (int nn = 0; nn < kN; ++nn) {
    a[nn] = -__expf(A_logs[(k * kD + d) * kN + nn]);
    P[nn] = 1.f;
    h[nn] = FINAL ? hpre_ws[((size_t)bks * kN + nn) * kD + d] : 0.f;
  }
  const float Dv = Dsv[k * kD + d];
  const size_t rowBase = (size_t)bk * kL;
  const int l_begin = s * kSeg;

  __shared__ float sB[16][kN];
  __shared__ float sC[16][kN];

  for (int l0 = l_begin; l0 < l_begin + kSeg; l0 += 16) {
    __syncthreads();
    for (int i = threadIdx.x; i < 16 * kN; i += 96) {
      const int t = i >> 4, nn = i & 15;
      sB[t][nn] = Bs_ws[(rowBase + l0 + t) * kN + nn];
      sC[t][nn] = Cs_ws[(rowBase + l0 + t) * kN + nn];
    }
    __syncthreads();
#pragma unroll 4
    for (int t = 0; t < 16; ++t) {
      const int l = l0 + t;
      const float dt = dt_ws[(rowBase + l) * kD + d];
      const int gl = k ? (kL - 1 - l) : l;
      const float u = x[((size_t)b * kL + gl) * kD + d];
      const float du = dt * u;
      float y = 0.f;
#pragma unroll
      for (int nn = 0; nn < kN; ++nn) {
        const float dA = __expf(dt * a[nn]);
        h[nn] = fmaf(dA, h[nn], du * sB[t][nn]);
        if (!FINAL) P[nn] *= dA;
        if (FINAL) y = fmaf(h[nn], sC[t][nn], y);
      }
      if (FINAL) ys_ws[(rowBase + l) * kD + d] = fmaf(u, Dv, y);
    }
  }

  if (!FINAL) {
#pragma unroll
    for (int nn = 0; nn < kN; ++nn) {
      P_ws[((size_t)bks * kN + nn) * kD + d] = P[nn];
      Q_ws[((size_t)bks * kN + nn) * kD + d] = h[nn];
    }
  }
}

// ---------------------------------------------------------------------------
// Kernel 2b: sequential combine across the 32 segments (tiny).
// Grid: B*K*N blocks of 96 threads (d fastest -> coalesced).
// ---------------------------------------------------------------------------
__global__ __launch_bounds__(96) void ss1d_scan_combine_kernel(
    const float* __restrict__ P_ws, const float* __restrict__ Q_ws,
    float* __restrict__ hpre_ws) {
  const int bkn = blockIdx.x;  // bk * kN + n
  const int bk = bkn >> 4;
  const int nn = bkn & 15;
  const int d = threadIdx.x;
  float hp = 0.f;
  for (int s = 0; s < kS; ++s) {
    const size_t idx = ((size_t)(bk * kS + s) * kN + nn) * kD + d;
    hpre_ws[idx] = hp;
    hp = fmaf(P_ws[idx], hp, Q_ws[idx]);
  }
}

// ---------------------------------------------------------------------------
// Kernel 3: merge forward pass + flipped backward pass.
// ---------------------------------------------------------------------------
__global__ __launch_bounds__(256) void ss1d_merge_kernel(
    const float* __restrict__ ys_ws, const float* __restrict__ merge_w,
    const float* __restrict__ merge_b, float* __restrict__ out) {
  const int idx = blockIdx.x * 256 + threadIdx.x;  // over B*L*D
  const int d = idx % kD;
  const int l = (idx / kD) % kL;
  const int b = idx / (kL * kD);
  const float w0 = merge_w[0];
  const float w1 = merge_w[1];
  const float mb = merge_b[0];
  const float y1 = ys_ws[(((size_t)(b * kK + 0)) * kL + l) * kD + d];
  const float y2 = ys_ws[(((size_t)(b * kK + 1)) * kL + (kL - 1 - l)) * kD + d];
  out[idx] = fmaf(w0, y1, fmaf(w1, y2, mb));
}

extern "C" void kernel_launch(void* const* d_in, const int* in_sizes, int n_in,
                              void* d_out, int out_size, void* d_ws,
                              size_t ws_size, hipStream_t stream) {
  const float* x = (const float*)d_in[0];    // (B, L, D)
  const float* xw = (const float*)d_in[1];   // (K, C, D)
  const float* dtw = (const float*)d_in[2];  // (K, D, R)
  const float* dtb = (const float*)d_in[3];  // (K, D)
  const float* Al = (const float*)d_in[4];   // (K*D, N)
  const float* Ds = (const float*)d_in[5];   // (K*D,)
  const float* mw = (const float*)d_in[6];   // (K,)
  const float* mb = (const float*)d_in[7];   // ()

  float* ws = (float*)d_ws;
  size_t off = 0;
  float* dt_ws = ws + off;   off += (size_t)kB * kK * kL * kD;      // 12.6 MB
  float* Bs_ws = ws + off;   off += (size_t)kB * kK * kL * kN;      //  2.1 MB
  float* Cs_ws = ws + off;   off += (size_t)kB * kK * kL * kN;      //  2.1 MB
  float* ys_ws = ws + off;   off += (size_t)kB * kK * kL * kD;      // 12.6 MB
  float* P_ws  = ws + off;   off += (size_t)kB * kK * kS * kN * kD; //  1.6 MB
  float* Q_ws  = ws + off;   off += (size_t)kB * kK * kS * kN * kD; //  1.6 MB
  float* Hp_ws = ws + off;   off += (size_t)kB * kK * kS * kN * kD; //  1.6 MB

  // 1) WMMA projection (x_dbl split + dt with softplus)
  ss1d_proj_kernel<<<dim3(kB * kK, kL / 128), 256, 0, stream>>>(
      x, xw, dtw, dtb, dt_ws, Bs_ws, Cs_ws);

  // 2) segmented scan: partials -> combine -> final
  ss1d_scan_seg_kernel<0><<<kB * kK * kS, 96, 0, stream>>>(
      x, Al, Ds, dt_ws, Bs_ws, Cs_ws, nullptr, P_ws, Q_ws, nullptr);
  ss1d_scan_combine_kernel<<<kB * kK * kN, 96, 0, stream>>>(P_ws, Q_ws, Hp_ws);
  ss1d_scan_seg_kernel<1><<<kB * kK * kS, 96, 0, stream>>>(
      x, Al, Ds, dt_ws, Bs_ws, Cs_ws, Hp_ws, nullptr, nullptr, ys_ws);

  // 3) merge the two directions
  ss1d_merge_kernel<<<(kB * kL * kD) / 256, 256, 0, stream>>>(
      ys_ws, mw, mb, (float*)d_out);
}